// hausdorff_loss_14491219656942
// MI455X (gfx1250) — compile-verified
//
#include <hip/hip_runtime.h>
#include <hip/hip_bf16.h>

#define BIGF 1.0e8f
#define NVOX (2 * 64 * 64 * 64)   // 524288 voxels

typedef float v2f __attribute__((ext_vector_type(2)));
typedef float v8f __attribute__((ext_vector_type(8)));

#if defined(__gfx1250__) && __has_builtin(__builtin_amdgcn_wmma_f32_16x16x4_f32)
#define HAVE_WMMA 1
#else
#define HAVE_WMMA 0
#endif

#if defined(__gfx1250__) && __has_builtin(__builtin_amdgcn_global_load_async_to_lds_b32)
#define HAVE_ASYNC 1
#else
#define HAVE_ASYNC 0
#endif

// f[idx] = BIG where (mmap in mask-set) XOR invert, else 0
__global__ void k_init_f(float* __restrict__ f, const int* __restrict__ mmap,
                         unsigned mask, int invert) {
  int idx = blockIdx.x * blockDim.x + threadIdx.x;
  if (idx >= NVOX) return;
  int v = mmap[idx];
  int member = (v >= 0 && v < 32) ? (int)((mask >> v) & 1u) : 0;
  f[idx] = (member ^ invert) ? BIGF : 0.0f;
}

// One exact 1-D squared-EDT sweep along an axis with stride 2^ls.
// 256 threads = 4 lines of 64; line staged in LDS (async on gfx1250).
__global__ void k_edt_sweep(float* __restrict__ f, int ls) {
  __shared__ float tile[4 * 64];
  const int tid = threadIdx.x;      // 0..255
  const int ll  = tid >> 6;         // line within block, 0..3
  const int i   = tid & 63;         // position within line
  const int stride  = 1 << ls;
  const int lineIdx = blockIdx.x * 4 + ll;
  const int base = (lineIdx >> ls) * (stride << 6) + (lineIdx & (stride - 1));
  const int addr = base + i * stride;

#if HAVE_ASYNC
  {
    int* gp = (int*)(f + addr);
    int* lp = (int*)(&tile[tid]);
    __builtin_amdgcn_global_load_async_to_lds_b32(
        (__attribute__((address_space(1))) int*)gp,
        (__attribute__((address_space(3))) int*)lp,
        0, 0);
  }
#if __has_builtin(__builtin_amdgcn_s_wait_asynccnt)
  __builtin_amdgcn_s_wait_asynccnt(0);
#else
  asm volatile("s_wait_asynccnt 0" ::: "memory");
#endif
#else
  tile[tid] = f[addr];
#endif
  __syncthreads();

  const float* row = &tile[ll << 6];
  float m = 3.0e38f;
#pragma unroll 16
  for (int j = 0; j < 64; ++j) {
    int d = i - j;
    m = fminf(m, row[j] + (float)(d * d));   // same f32 add/min as reference
  }
  f[addr] = m;
}

// bnd = (sqB == 1); f (holding sqB) := BIG where (sqA==0 && bnd==0) else 0
__global__ void k_bnd_lvl0(float* __restrict__ bnd, float* __restrict__ f_sqB,
                           const float* __restrict__ sqA) {
  int idx = blockIdx.x * blockDim.x + threadIdx.x;
  if (idx >= NVOX) return;
  float sb = f_sqB[idx];
  float b = (sb == 1.0f) ? 1.0f : 0.0f;
  bnd[idx] = b;
  float sa = sqA[idx];
  f_sqB[idx] = (sa == 0.0f && b == 0.0f) ? BIGF : 0.0f;
}

// tsd = sqrt(sqA) - sqrt(sqC)
__global__ void k_tsd(float* __restrict__ tsd, const float* __restrict__ sqC,
                      const float* __restrict__ sqA) {
  int idx = blockIdx.x * blockDim.x + threadIdx.x;
  if (idx >= NVOX) return;
  tsd[idx] = sqrtf(sqA[idx]) - sqrtf(sqC[idx]);
}

// Stage-1 reduction: 512 blocks x 1024 elements. Sums via WMMA ones-matmul
// (D = A x ones + C  =>  sum(D) = 16 * sum(A)); maxes via LDS tree.
__global__ void k_reduce1(const float* __restrict__ bndA, const float* __restrict__ tsdA,
                          const float* __restrict__ bndB, const float* __restrict__ tsdB,
                          float* __restrict__ partials) {
  const int blk  = blockIdx.x;    // 0..511
  const int tid  = threadIdx.x;   // 0..255
  const int wv   = tid >> 5;      // wave32
  const int lane = tid & 31;
  const int base = blk * 1024 + wv * 128;

  float m1 = 0.0f, m3 = 0.0f;
#if HAVE_WMMA
  v8f c0 = {0,0,0,0,0,0,0,0}, c1 = {0,0,0,0,0,0,0,0};
  v8f c2 = {0,0,0,0,0,0,0,0}, c3 = {0,0,0,0,0,0,0,0};
  const v2f ones = {1.0f, 1.0f};
#else
  float s0 = 0.0f, s1 = 0.0f, s2 = 0.0f, s3 = 0.0f;
#endif
  for (int it = 0; it < 2; ++it) {
    const int e0 = base + it * 64 + lane;
    const int e1 = e0 + 32;
    float ba0 = bndA[e0], ba1 = bndA[e1];
    float bb0 = bndB[e0], bb1 = bndB[e1];
    float ta0 = tsdA[e0], ta1 = tsdA[e1];
    float tb0 = tsdB[e0], tb1 = tsdB[e1];
    float hi0 = fabsf(ba0 * tb0), hi1 = fabsf(ba1 * tb1);
    float ho0 = fabsf(bb0 * ta0), ho1 = fabsf(bb1 * ta1);
    m1 = fmaxf(m1, fmaxf(hi0, hi1));
    m3 = fmaxf(m3, fmaxf(ho0, ho1));
#if HAVE_WMMA
    v2f a;
    a[0] = ba0; a[1] = ba1;
    c0 = __builtin_amdgcn_wmma_f32_16x16x4_f32(false, a, false, ones, (short)0, c0, false, false);
    a[0] = hi0; a[1] = hi1;
    c1 = __builtin_amdgcn_wmma_f32_16x16x4_f32(false, a, false, ones, (short)0, c1, false, false);
    a[0] = bb0; a[1] = bb1;
    c2 = __builtin_amdgcn_wmma_f32_16x16x4_f32(false, a, false, ones, (short)0, c2, false, false);
    a[0] = ho0; a[1] = ho1;
    c3 = __builtin_amdgcn_wmma_f32_16x16x4_f32(false, a, false, ones, (short)0, c3, false, false);
#else
    s0 += ba0 + ba1; s1 += hi0 + hi1; s2 += bb0 + bb1; s3 += ho0 + ho1;
#endif
  }
#if HAVE_WMMA
  float s0 = 0.0f, s1 = 0.0f, s2 = 0.0f, s3 = 0.0f;
#pragma unroll
  for (int r = 0; r < 8; ++r) { s0 += c0[r]; s1 += c1[r]; s2 += c2[r]; s3 += c3[r]; }
  const float SC = 1.0f / 16.0f;   // sum(D) = 16 * sum(A)
#else
  const float SC = 1.0f;
#endif

  __shared__ float red[256];
  float vals[6] = {s0, s1, s2, s3, m1, m3};
  for (int q = 0; q < 6; ++q) {
    red[tid] = vals[q];
    __syncthreads();
    for (int off = 128; off > 0; off >>= 1) {
      if (tid < off)
        red[tid] = (q < 4) ? (red[tid] + red[tid + off]) : fmaxf(red[tid], red[tid + off]);
      __syncthreads();
    }
    if (tid == 0) partials[q * 512 + blk] = (q < 4) ? red[0] * SC : red[0];
    __syncthreads();
  }
}

// Stage-2: single block folds 512 partials per quantity, emits the 5 outputs.
__global__ void k_reduce2(const float* __restrict__ partials, float* __restrict__ out) {
  __shared__ float red[256];
  __shared__ float res[6];
  const int tid = threadIdx.x;   // 0..255
  for (int q = 0; q < 6; ++q) {
    float a = partials[q * 512 + tid];
    float b = partials[q * 512 + 256 + tid];
    red[tid] = (q < 4) ? (a + b) : fmaxf(a, b);
    __syncthreads();
    for (int off = 128; off > 0; off >>= 1) {
      if (tid < off)
        red[tid] = (q < 4) ? (red[tid] + red[tid + off]) : fmaxf(red[tid], red[tid + off]);
      __syncthreads();
    }
    if (tid == 0) res[q] = red[0];
    __syncthreads();
  }
  if (tid == 0) {
    float him = res[1] / res[0];            // h_in_mean
    float hom = res[3] / res[2];            // h_out_mean
    float da = him - 2.0f, db = hom - 2.0f;
    out[0] = him;
    out[1] = res[4];                        // hausdorff_inner.max()
    out[2] = hom;
    out[3] = res[5];                        // hausdorff_outer.max()
    out[4] = da * da + db * db;             // loss
  }
}

extern "C" void kernel_launch(void* const* d_in, const int* in_sizes, int n_in,
                              void* d_out, int out_size, void* d_ws, size_t ws_size,
                              hipStream_t stream) {
  const int* mmap = (const int*)d_in[0];
  float* out = (float*)d_out;

  float* V0       = (float*)d_ws;          // working EDT volume (sqB -> lvl0 -> sqC)
  float* V1       = V0 + NVOX;             // sqA
  float* bndA     = V1 + NVOX;
  float* tsdA     = bndA + NVOX;
  float* bndB     = tsdA + NVOX;
  float* tsdB     = bndB + NVOX;
  float* partials = tsdB + NVOX;           // 6 * 512 floats

  const dim3 eb(256);
  const dim3 eg(NVOX / 256);               // 2048 blocks elementwise
  const dim3 sg(8192 / 4);                 // 2048 blocks, 4 lines each

  for (int cfg = 0; cfg < 2; ++cfg) {
    // cfgA: lz = mmap in {0,1,2,4} (mask 0x17); cfgB: {0,1,4} (mask 0x13)
    const unsigned mask = (cfg == 0) ? 0x17u : 0x13u;
    float* bnd = (cfg == 0) ? bndA : bndB;
    float* tsd = (cfg == 0) ? tsdA : tsdB;

    // sqA = EDT^2(lz) into V1
    k_init_f<<<eg, eb, 0, stream>>>(V1, mmap, mask, 0);
    k_edt_sweep<<<sg, eb, 0, stream>>>(V1, 0);    // z (stride 1)
    k_edt_sweep<<<sg, eb, 0, stream>>>(V1, 6);    // y (stride 64)
    k_edt_sweep<<<sg, eb, 0, stream>>>(V1, 12);   // x (stride 4096)

    // sqB = EDT^2(1-lz) into V0
    k_init_f<<<eg, eb, 0, stream>>>(V0, mmap, mask, 1);
    k_edt_sweep<<<sg, eb, 0, stream>>>(V0, 0);
    k_edt_sweep<<<sg, eb, 0, stream>>>(V0, 6);
    k_edt_sweep<<<sg, eb, 0, stream>>>(V0, 12);

    // boundary + level_zero_inner seed, then sqC = EDT^2(lvl0) into V0
    k_bnd_lvl0<<<eg, eb, 0, stream>>>(bnd, V0, V1);
    k_edt_sweep<<<sg, eb, 0, stream>>>(V0, 0);
    k_edt_sweep<<<sg, eb, 0, stream>>>(V0, 6);
    k_edt_sweep<<<sg, eb, 0, stream>>>(V0, 12);

    k_tsd<<<eg, eb, 0, stream>>>(tsd, V0, V1);
  }

  k_reduce1<<<512, 256, 0, stream>>>(bndA, tsdA, bndB, tsdB, partials);
  k_reduce2<<<1, 256, 0, stream>>>(partials, out);
}